// SeparateTrackLayer_16226386444313
// MI455X (gfx1250) — compile-verified
//
#include <hip/hip_runtime.h>
#include <hip/hip_bf16.h>

#define C_DIM 128
#define TILE_M 16
#define BN_EPS 1e-5f

typedef float v2f __attribute__((ext_vector_type(2)));
typedef float v8f __attribute__((ext_vector_type(8)));

// ---------------------------------------------------------------------------
// h_in = (h_local + h_global[max(node_ids,0)]) * valid       (float4 per thread)
// ---------------------------------------------------------------------------
__global__ void gather_kernel(const float* __restrict__ h_local,
                              const float* __restrict__ h_global,
                              const int* __restrict__ node_ids,
                              const unsigned char* __restrict__ valid,
                              float* __restrict__ h_in, int M) {
    int gid = blockIdx.x * blockDim.x + threadIdx.x;
    int total = M * (C_DIM / 4);
    if (gid >= total) return;
    int row = gid >> 5;          // C_DIM/4 == 32
    int c4  = gid & 31;
    int nid = node_ids[row]; if (nid < 0) nid = 0;
    float vf = valid[row] ? 1.0f : 0.0f;
    float4 a = ((const float4*)(h_local  + (size_t)row * C_DIM))[c4];
    float4 b = ((const float4*)(h_global + (size_t)nid * C_DIM))[c4];
    float4 r;
    r.x = (a.x + b.x) * vf; r.y = (a.y + b.y) * vf;
    r.z = (a.z + b.z) * vf; r.w = (a.w + b.w) * vf;
    ((float4*)(h_in + (size_t)row * C_DIM))[c4] = r;
}

// ---------------------------------------------------------------------------
// Edge aggregation: one wave per edge, lane handles 4 channels (float4 load,
// 4x global_atomic_add_f32 scatter).  agg[dst] += h_in[src]
// ---------------------------------------------------------------------------
__global__ void edge_agg_kernel(const int* __restrict__ src,
                                const int* __restrict__ dst,
                                const float* __restrict__ h_in,
                                float* __restrict__ agg, int E) {
    int gid = blockIdx.x * blockDim.x + threadIdx.x;
    int e = gid >> 5;
    if (e >= E) return;
    int lane = threadIdx.x & 31;
    int s = src[e];
    int d = dst[e];
    float4 v = ((const float4*)(h_in + (size_t)s * C_DIM))[lane];
    float* ap = agg + (size_t)d * C_DIM + lane * 4;
    atomicAdd(ap + 0, v.x);
    atomicAdd(ap + 1, v.y);
    atomicAdd(ap + 2, v.z);
    atomicAdd(ap + 3, v.w);
}

// ---------------------------------------------------------------------------
// Fused GIN MLP:  out = relu(((1+eps)*h_in + agg) @ W1 + b1) @ W2 + b2, *valid
// 256 threads (8 waves) per 16-row tile.  Uses V_WMMA_F32_16X16X4_F32.
//
// Fragment layouts (wave32, f32):
//   A 16x4 : lane m = lane&15, v2f elem r at K = kbase + 2*(lane>>4) + r
//   B 4x16 : col n = lane&15,  v2f elem r at K = kbase + 2*(lane>>4) + r
//   C/D    : v8f elem r -> row (r + 8*(lane>>4)), col n
//
// Weights are stored K-pair-interleaved in LDS:
//   sW[(k>>1)*2*C + col*2 + (k&1)]  ->  B fragment = one contiguous ds_load_b64
// (ka = kbase + 2h is always even, so (ka, ka+1) is one stored pair.)
// ---------------------------------------------------------------------------
__global__ __launch_bounds__(256) void gin_mlp_kernel(
    const float* __restrict__ h_in, const float* __restrict__ agg,
    const float* __restrict__ W1, const float* __restrict__ b1,
    const float* __restrict__ W2, const float* __restrict__ b2,
    const float* __restrict__ eps_p, const unsigned char* __restrict__ valid,
    float* __restrict__ out, int M) {
    __shared__ float sW1[C_DIM * C_DIM];   // 64 KB, pair-interleaved
    __shared__ float sW2[C_DIM * C_DIM];   // 64 KB, pair-interleaved
    __shared__ float sB1[C_DIM];
    __shared__ float sB2[C_DIM];
    __shared__ float sIn[TILE_M * C_DIM];  // 8 KB
    __shared__ float sMid[TILE_M * C_DIM]; // 8 KB
    __shared__ float sVal[TILE_M];

    const int tid  = threadIdx.x;
    const int row0 = blockIdx.x * TILE_M;
    const float opE = 1.0f + eps_p[0];

    // cooperative load of weights (pair-interleaved) & biases
    {
        const float4* w1v = (const float4*)W1;
        const float4* w2v = (const float4*)W2;
        for (int i = tid; i < C_DIM * C_DIM / 4; i += 256) {
            int k  = i >> 5;            // row of W
            int c0 = (i & 31) * 4;      // first of 4 consecutive cols
            float4 v1 = w1v[i];
            float4 v2 = w2v[i];
            int base = (k >> 1) * (2 * C_DIM) + (k & 1);
            sW1[base + (c0 + 0) * 2] = v1.x;
            sW1[base + (c0 + 1) * 2] = v1.y;
            sW1[base + (c0 + 2) * 2] = v1.z;
            sW1[base + (c0 + 3) * 2] = v1.w;
            sW2[base + (c0 + 0) * 2] = v2.x;
            sW2[base + (c0 + 1) * 2] = v2.y;
            sW2[base + (c0 + 2) * 2] = v2.z;
            sW2[base + (c0 + 3) * 2] = v2.w;
        }
        if (tid < C_DIM) { sB1[tid] = b1[tid]; sB2[tid] = b2[tid]; }
        if (tid < TILE_M) {
            int r = row0 + tid;
            sVal[tid] = (r < M && valid[r]) ? 1.0f : 0.0f;
        }
    }
    // input tile: pre = (1+eps)*h_in + agg
    for (int i = tid; i < TILE_M * C_DIM / 4; i += 256) {
        int r  = i >> 5;   // / (C_DIM/4)
        int c4 = i & 31;
        int row = row0 + r; if (row >= M) row = M - 1;
        float4 hv = ((const float4*)(h_in + (size_t)row * C_DIM))[c4];
        float4 av = ((const float4*)(agg  + (size_t)row * C_DIM))[c4];
        float4 p;
        p.x = opE * hv.x + av.x; p.y = opE * hv.y + av.y;
        p.z = opE * hv.z + av.z; p.w = opE * hv.w + av.w;
        ((float4*)(sIn + r * C_DIM))[c4] = p;
    }
    __syncthreads();

    const int wave = tid >> 5;
    const int lane = tid & 31;
    const int n    = lane & 15;
    const int hh   = lane >> 4;
    const int col  = wave * 16 + n;

    // ---- GEMM1 + bias + ReLU -> sMid
    v8f acc;
#pragma unroll
    for (int r = 0; r < 8; ++r) acc[r] = sB1[col];
#pragma unroll
    for (int k = 0; k < C_DIM; k += 4) {
        const int ka = k + 2 * hh;   // even
        v2f a = *(const v2f*)(sIn + n * C_DIM + ka);
        v2f b = *(const v2f*)(sW1 + (ka >> 1) * (2 * C_DIM) + col * 2);
        acc = __builtin_amdgcn_wmma_f32_16x16x4_f32(false, a, false, b,
                                                    (short)0, acc, false, false);
    }
#pragma unroll
    for (int r = 0; r < 8; ++r)
        sMid[(r + 8 * hh) * C_DIM + col] = acc[r] > 0.0f ? acc[r] : 0.0f;
    __syncthreads();

    // ---- GEMM2 + bias + valid mask -> out (pre-BN h_local_new)
    v8f acc2;
#pragma unroll
    for (int r = 0; r < 8; ++r) acc2[r] = sB2[col];
#pragma unroll
    for (int k = 0; k < C_DIM; k += 4) {
        const int ka = k + 2 * hh;
        v2f a = *(const v2f*)(sMid + n * C_DIM + ka);
        v2f b = *(const v2f*)(sW2 + (ka >> 1) * (2 * C_DIM) + col * 2);
        acc2 = __builtin_amdgcn_wmma_f32_16x16x4_f32(false, a, false, b,
                                                     (short)0, acc2, false, false);
    }
#pragma unroll
    for (int r = 0; r < 8; ++r) {
        int m = r + 8 * hh;
        int row = row0 + m;
        if (row < M)
            out[(size_t)row * C_DIM + col] = acc2[r] * sVal[m];
    }
}

// ---------------------------------------------------------------------------
// Scatter-mean prep: gsum[nid] += h2[i], vcnt[nid] += 1  (one wave per node)
// ---------------------------------------------------------------------------
__global__ void scatter_global_kernel(const float* __restrict__ h2,
                                      const int* __restrict__ node_ids,
                                      const unsigned char* __restrict__ valid,
                                      float* __restrict__ gsum,
                                      float* __restrict__ vcnt, int M) {
    int gid = blockIdx.x * blockDim.x + threadIdx.x;
    int i = gid >> 5;
    if (i >= M) return;
    if (!valid[i]) return;
    int lane = threadIdx.x & 31;
    int g = node_ids[i]; if (g < 0) g = 0;
    float4 v = ((const float4*)(h2 + (size_t)i * C_DIM))[lane];
    float* gp = gsum + (size_t)g * C_DIM + lane * 4;
    atomicAdd(gp + 0, v.x);
    atomicAdd(gp + 1, v.y);
    atomicAdd(gp + 2, v.z);
    atomicAdd(gp + 3, v.w);
    if (lane == 0) atomicAdd(vcnt + g, 1.0f);
}

// g[row] /= max(vcnt[row], 1)   (in place)
__global__ void gdiv_kernel(float* __restrict__ g, const float* __restrict__ vcnt, int N) {
    int gid = blockIdx.x * blockDim.x + threadIdx.x;
    int total = N * (C_DIM / 4);
    if (gid >= total) return;
    int row = gid >> 5;
    int c4  = gid & 31;
    float cnt = vcnt[row]; if (cnt < 1.0f) cnt = 1.0f;
    float inv = 1.0f / cnt;
    float4 x = ((float4*)(g + (size_t)row * C_DIM))[c4];
    x.x *= inv; x.y *= inv; x.z *= inv; x.w *= inv;
    ((float4*)(g + (size_t)row * C_DIM))[c4] = x;
}

// ---------------------------------------------------------------------------
// Column stats: thread = channel, block covers rows_per_block rows.
// ---------------------------------------------------------------------------
__global__ void colstats_kernel(const float* __restrict__ x, int rows,
                                int rows_per_block,
                                float* __restrict__ sumbuf,
                                float* __restrict__ sqbuf) {
    int c  = threadIdx.x;                  // 128 threads
    int r0 = blockIdx.x * rows_per_block;
    int r1 = r0 + rows_per_block; if (r1 > rows) r1 = rows;
    float s = 0.0f, q = 0.0f;
    for (int r = r0; r < r1; ++r) {
        float v = x[(size_t)r * C_DIM + c];
        s += v; q += v * v;
    }
    atomicAdd(&sumbuf[c], s);
    atomicAdd(&sqbuf[c], q);
}

// stats layout: [sum_l, sq_l, sum_g, sq_g] in; [scale_l, shift_l, scale_g, shift_g] out
__global__ void finalize_stats_kernel(const float* __restrict__ sums,
                                      const float* __restrict__ gamma_l,
                                      const float* __restrict__ beta_l,
                                      const float* __restrict__ gamma_g,
                                      const float* __restrict__ beta_g,
                                      float* __restrict__ scales, int M, int N) {
    int t = threadIdx.x;  // 256
    if (t < C_DIM) {
        float cnt = (float)M;
        float mu  = sums[t] / cnt;
        float var = sums[C_DIM + t] / cnt - mu * mu;
        float sc  = gamma_l[t] * rsqrtf(var + BN_EPS);
        scales[t]         = sc;
        scales[C_DIM + t] = beta_l[t] - mu * sc;
    } else {
        int c = t - C_DIM;
        float cnt = (float)N;
        float mu  = sums[2 * C_DIM + c] / cnt;
        float var = sums[3 * C_DIM + c] / cnt - mu * mu;
        float sc  = gamma_g[c] * rsqrtf(var + BN_EPS);
        scales[2 * C_DIM + c] = sc;
        scales[3 * C_DIM + c] = beta_g[c] - mu * sc;
    }
}

// out = (h2*scale + shift + h_local) * valid     (in place on d_out region 1)
__global__ void final_local_kernel(float* __restrict__ out,
                                   const float* __restrict__ h_local,
                                   const unsigned char* __restrict__ valid,
                                   const float* __restrict__ ss, int M) {
    int gid = blockIdx.x * blockDim.x + threadIdx.x;
    int total = M * (C_DIM / 4);
    if (gid >= total) return;
    int row = gid >> 5;
    int c4  = gid & 31;
    float vf = valid[row] ? 1.0f : 0.0f;
    float4 x  = ((float4*)(out + (size_t)row * C_DIM))[c4];
    float4 sc = ((const float4*)ss)[c4];
    float4 sh = ((const float4*)(ss + C_DIM))[c4];
    float4 hl = ((const float4*)(h_local + (size_t)row * C_DIM))[c4];
    float4 r;
    r.x = (x.x * sc.x + sh.x + hl.x) * vf;
    r.y = (x.y * sc.y + sh.y + hl.y) * vf;
    r.z = (x.z * sc.z + sh.z + hl.z) * vf;
    r.w = (x.w * sc.w + sh.w + hl.w) * vf;
    ((float4*)(out + (size_t)row * C_DIM))[c4] = r;
}

// out = g*scale + shift + h_global               (in place on d_out region 2)
__global__ void final_global_kernel(float* __restrict__ out,
                                    const float* __restrict__ h_global,
                                    const float* __restrict__ ss, int N) {
    int gid = blockIdx.x * blockDim.x + threadIdx.x;
    int total = N * (C_DIM / 4);
    if (gid >= total) return;
    int row = gid >> 5;
    int c4  = gid & 31;
    float4 x  = ((float4*)(out + (size_t)row * C_DIM))[c4];
    float4 sc = ((const float4*)ss)[c4];
    float4 sh = ((const float4*)(ss + C_DIM))[c4];
    float4 hg = ((const float4*)(h_global + (size_t)row * C_DIM))[c4];
    float4 r;
    r.x = x.x * sc.x + sh.x + hg.x;
    r.y = x.y * sc.y + sh.y + hg.y;
    r.z = x.z * sc.z + sh.z + hg.z;
    r.w = x.w * sc.w + sh.w + hg.w;
    ((float4*)(out + (size_t)row * C_DIM))[c4] = r;
}

// ---------------------------------------------------------------------------
extern "C" void kernel_launch(void* const* d_in, const int* in_sizes, int n_in,
                              void* d_out, int out_size, void* d_ws, size_t ws_size,
                              hipStream_t stream) {
    const float* h_local  = (const float*)d_in[0];
    const float* h_global = (const float*)d_in[1];
    // d_in[2] ea_flat: unused by 'gin'
    const float* W1  = (const float*)d_in[3];
    const float* b1  = (const float*)d_in[4];
    const float* W2  = (const float*)d_in[5];
    const float* b2  = (const float*)d_in[6];
    const float* eps = (const float*)d_in[7];
    const float* gamma_l = (const float*)d_in[8];
    const float* beta_l  = (const float*)d_in[9];
    const float* gamma_g = (const float*)d_in[10];
    const float* beta_g  = (const float*)d_in[11];
    const int*   ei      = (const int*)d_in[12];
    const int*   node_ids = (const int*)d_in[13];
    const unsigned char* valid = (const unsigned char*)d_in[14];  // jnp.bool_ = 1 byte

    const int M = in_sizes[0] / C_DIM;
    const int N = in_sizes[1] / C_DIM;
    const int E = in_sizes[12] / 2;
    const int* src = ei;
    const int* dst = ei + E;

    // workspace layout (floats)
    float* ws    = (float*)d_ws;
    float* h_in  = ws;                                  // M*C
    float* agg   = h_in + (size_t)M * C_DIM;            // M*C
    float* vcnt  = agg  + (size_t)M * C_DIM;            // N
    float* stats = vcnt + (size_t)N;                    // 512
    float* scales = stats + 4 * C_DIM;                  // [scale_l,shift_l,scale_g,shift_g]

    float* out_l = (float*)d_out;                       // M*C (h2, then final local)
    float* out_g = out_l + (size_t)M * C_DIM;           // N*C (gsum, then final global)

    // zero accumulators (graph-capturable async memsets)
    hipMemsetAsync(agg,  0, (size_t)M * C_DIM * sizeof(float), stream);
    hipMemsetAsync(vcnt, 0, (size_t)N * sizeof(float), stream);
    hipMemsetAsync(stats, 0, 4 * C_DIM * sizeof(float), stream);
    hipMemsetAsync(out_g, 0, (size_t)N * C_DIM * sizeof(float), stream);

    // 1) h_in = (h_local + h_global[nid]) * valid
    {
        long long total = (long long)M * (C_DIM / 4);
        int blocks = (int)((total + 255) / 256);
        gather_kernel<<<blocks, 256, 0, stream>>>(h_local, h_global, node_ids, valid, h_in, M);
    }
    // 2) edge aggregation
    {
        long long total = (long long)E * 32;
        int blocks = (int)((total + 255) / 256);
        edge_agg_kernel<<<blocks, 256, 0, stream>>>(src, dst, h_in, agg, E);
    }
    // 3) fused GIN MLP (WMMA f32) -> out_l holds masked h2
    {
        int blocks = (M + TILE_M - 1) / TILE_M;
        gin_mlp_kernel<<<blocks, 256, 0, stream>>>(h_in, agg, W1, b1, W2, b2, eps, valid, out_l, M);
    }
    // 4) scatter into global sums
    {
        long long total = (long long)M * 32;
        int blocks = (int)((total + 255) / 256);
        scatter_global_kernel<<<blocks, 256, 0, stream>>>(out_l, node_ids, valid, out_g, vcnt, M);
    }
    // 5) divide -> pre-BN h_global_new (in place)
    {
        long long total = (long long)N * (C_DIM / 4);
        int blocks = (int)((total + 255) / 256);
        gdiv_kernel<<<blocks, 256, 0, stream>>>(out_g, vcnt, N);
    }
    // 6) column stats (local + global)
    {
        const int RPB = 512;
        colstats_kernel<<<(M + RPB - 1) / RPB, C_DIM, 0, stream>>>(out_l, M, RPB, stats, stats + C_DIM);
        colstats_kernel<<<(N + RPB - 1) / RPB, C_DIM, 0, stream>>>(out_g, N, RPB, stats + 2 * C_DIM, stats + 3 * C_DIM);
    }
    // 7) BN scale/shift
    finalize_stats_kernel<<<1, 256, 0, stream>>>(stats, gamma_l, beta_l, gamma_g, beta_g, scales, M, N);
    // 8) epilogues (in place on d_out)
    {
        long long total = (long long)M * (C_DIM / 4);
        int blocks = (int)((total + 255) / 256);
        final_local_kernel<<<blocks, 256, 0, stream>>>(out_l, h_local, valid, scales, M);
    }
    {
        long long total = (long long)N * (C_DIM / 4);
        int blocks = (int)((total + 255) / 256);
        final_global_kernel<<<blocks, 256, 0, stream>>>(out_g, h_global, scales + 2 * C_DIM, N);
    }
}